// Autoencoder_70042326663889
// MI455X (gfx1250) — compile-verified
//
#include <hip/hip_runtime.h>

typedef float v2f __attribute__((ext_vector_type(2)));
typedef float v8f __attribute__((ext_vector_type(8)));

#define NROWS 10000
#define NEDGES 320000
#define MTILES 625   // 10000 / 16

// ---------------- zero accumulators (h accumulator + means/stds region) ----
__global__ void zero2_kernel(float* __restrict__ a, float* __restrict__ b, int n) {
    int t = blockIdx.x * blockDim.x + threadIdx.x;
    if (t < n) { a[t] = 0.0f; b[t] = 0.0f; }
}

// ---------------- xw0 = x[10000,512] @ w0[512,32] via f32 WMMA 16x16x4 -----
// One wave per 16-row strip; computes both 16-col output tiles sharing A.
__global__ void gemm_xw0_kernel(const float* __restrict__ x,
                                const float* __restrict__ w0,
                                float* __restrict__ xw0) {
    int wave = (blockIdx.x * blockDim.x + threadIdx.x) >> 5;
    if (wave >= MTILES) return;            // wave-uniform: EXEC stays all-ones
    int lane = threadIdx.x & 31;
    int half = lane >> 4;                  // 0 or 1
    int m    = lane & 15;

    const float* arow = x + (size_t)(wave * 16 + m) * 512 + 2 * half;

    v8f c0 = {};
    v8f c1 = {};
    #pragma unroll 4
    for (int k0 = 0; k0 < 512; k0 += 4) {
        v2f a = *(const v2f*)(arow + k0);          // A[m, k0+2h .. k0+2h+1]
        int kb = (k0 + 2 * half) * 32;
        v2f b0, b1;
        b0.x = w0[kb + m];        b0.y = w0[kb + 32 + m];        // cols 0..15
        b1.x = w0[kb + 16 + m];   b1.y = w0[kb + 48 + m];        // cols 16..31
        c0 = __builtin_amdgcn_wmma_f32_16x16x4_f32(false, a, false, b0,
                                                   (short)0, c0, false, false);
        c1 = __builtin_amdgcn_wmma_f32_16x16x4_f32(false, a, false, b1,
                                                   (short)0, c1, false, false);
    }
    float* o = xw0 + (size_t)(wave * 16 + 8 * half) * 32 + m;
    #pragma unroll
    for (int v = 0; v < 8; v++) {
        o[(size_t)v * 32]      = c0[v];
        o[(size_t)v * 32 + 16] = c1[v];
    }
}

// ---------------- h_acc[row] += val * xw0[col]  (32 lanes = 32 features) ---
__global__ void spmm1_kernel(const int* __restrict__ row, const int* __restrict__ col,
                             const float* __restrict__ val, const float* __restrict__ xw0,
                             float* __restrict__ hacc) {
    int t = blockIdx.x * blockDim.x + threadIdx.x;
    int e = t >> 5;
    if (e >= NEDGES) return;
    int lane = t & 31;
    int r = row[e];
    int c = col[e];
    float v = val[e];
    atomicAdd(&hacc[(size_t)r * 32 + lane], v * xw0[(size_t)c * 32 + lane]);
}

// ---------------- h = relu(h_acc);  hm = h@w_mean, hs = h@w_std (K=32) -----
__global__ void mid_kernel(const float* __restrict__ hacc,
                           const float* __restrict__ wmean,
                           const float* __restrict__ wstd,
                           float* __restrict__ hm, float* __restrict__ hs) {
    int t = blockIdx.x * blockDim.x + threadIdx.x;
    if (t >= NROWS * 16) return;
    int nrow = t >> 4;
    int j    = t & 15;
    const float* hrow = hacc + (size_t)nrow * 32;
    float sm = 0.0f, ss = 0.0f;
    #pragma unroll
    for (int k = 0; k < 32; k++) {
        float hv = fmaxf(hrow[k], 0.0f);
        sm = fmaf(hv, wmean[k * 16 + j], sm);
        ss = fmaf(hv, wstd[k * 16 + j], ss);
    }
    hm[t] = sm;
    hs[t] = ss;
}

// ---------------- means/stds scatter: lanes 0-15 -> mean, 16-31 -> std -----
__global__ void spmm2_kernel(const int* __restrict__ row, const int* __restrict__ col,
                             const float* __restrict__ val,
                             const float* __restrict__ hm, const float* __restrict__ hs,
                             float* __restrict__ means, float* __restrict__ stds) {
    int t = blockIdx.x * blockDim.x + threadIdx.x;
    int e = t >> 5;
    if (e >= NEDGES) return;
    int lane = t & 31;
    int r = row[e];
    int c = col[e];
    float v = val[e];
    if (lane < 16) {
        atomicAdd(&means[(size_t)r * 16 + lane], v * hm[(size_t)c * 16 + lane]);
    } else {
        int j = lane - 16;
        atomicAdd(&stds[(size_t)r * 16 + j], v * hs[(size_t)c * 16 + j]);
    }
}

// ---------------- z = means + eps * exp(stds) ------------------------------
__global__ void z_kernel(const float* __restrict__ means, const float* __restrict__ stds,
                         const float* __restrict__ eps, float* __restrict__ z) {
    int t = blockIdx.x * blockDim.x + threadIdx.x;
    if (t >= NROWS * 16) return;
    z[t] = means[t] + eps[t] * expf(stds[t]);
}

// ---------------- recon = z @ z^T via f32 WMMA, non-temporal stores --------
// grid: (ceil(625/8), 625); wave -> j-tile, blockIdx.y -> i-tile
__global__ void decoder_kernel(const float* __restrict__ z, float* __restrict__ out) {
    int jt = blockIdx.x * 8 + (threadIdx.x >> 5);
    if (jt >= MTILES) return;              // wave-uniform guard
    int it   = blockIdx.y;
    int lane = threadIdx.x & 31;
    int half = lane >> 4;
    int m    = lane & 15;

    const float* za = z + (size_t)(it * 16 + m) * 16 + 2 * half;  // A: rows i
    const float* zb = z + (size_t)(jt * 16 + m) * 16 + 2 * half;  // B: rows j (B=Z^T tile)

    v8f c = {};
    #pragma unroll
    for (int k0 = 0; k0 < 16; k0 += 4) {
        v2f a = *(const v2f*)(za + k0);
        v2f b = *(const v2f*)(zb + k0);
        c = __builtin_amdgcn_wmma_f32_16x16x4_f32(false, a, false, b,
                                                  (short)0, c, false, false);
    }
    float* o = out + (size_t)(it * 16 + 8 * half) * 10000 + (size_t)jt * 16 + m;
    #pragma unroll
    for (int v = 0; v < 8; v++) {
        __builtin_nontemporal_store(c[v], o + (size_t)v * 10000);
    }
}

extern "C" void kernel_launch(void* const* d_in, const int* in_sizes, int n_in,
                              void* d_out, int out_size, void* d_ws, size_t ws_size,
                              hipStream_t stream) {
    const float* x       = (const float*)d_in[0];
    const float* w0      = (const float*)d_in[1];
    const float* w_mean  = (const float*)d_in[2];
    const float* w_std   = (const float*)d_in[3];
    const float* adj_val = (const float*)d_in[4];
    const float* eps     = (const float*)d_in[5];
    const int*   adj_row = (const int*)d_in[6];
    const int*   adj_col = (const int*)d_in[7];

    float* out   = (float*)d_out;
    float* recon = out;                         // 100,000,000 floats
    float* means = out + 100000000;             // 160,000 floats
    float* stds  = means + 160000;              // 160,000 floats (contiguous after means)

    float* ws   = (float*)d_ws;
    float* xw0  = ws;                           // 320,000 floats
    float* hacc = ws + 320000;                  // 320,000 floats
    float* hm   = ws + 640000;                  // 160,000 floats
    float* hs   = ws + 800000;                  // 160,000 floats
    float* zbuf = ws + 960000;                  // 160,000 floats

    // zero h accumulator and the means+stds output region (atomic targets)
    zero2_kernel<<<1250, 256, 0, stream>>>(hacc, means, 320000);

    // xw0 = x @ w0   (625 waves, 8 waves/block)
    gemm_xw0_kernel<<<79, 256, 0, stream>>>(x, w0, xw0);

    // h_acc = A @ xw0  (one wave per edge)
    spmm1_kernel<<<40000, 256, 0, stream>>>(adj_row, adj_col, adj_val, xw0, hacc);

    // hm = relu(h) @ w_mean, hs = relu(h) @ w_std
    mid_kernel<<<625, 256, 0, stream>>>(hacc, w_mean, w_std, hm, hs);

    // means = A @ hm, stds = A @ hs
    spmm2_kernel<<<40000, 256, 0, stream>>>(adj_row, adj_col, adj_val, hm, hs, means, stds);

    // z = means + eps * exp(stds)
    z_kernel<<<625, 256, 0, stream>>>(means, stds, eps, zbuf);

    // recon = z @ z^T  (store-bandwidth bound: 400 MB of NT stores)
    dim3 dgrid(79, 625);
    decoder_kernel<<<dgrid, 256, 0, stream>>>(zbuf, recon);
}